// WeightedPolynormerLocal_59596966199880
// MI455X (gfx1250) — compile-verified
//
#include <hip/hip_runtime.h>
#include <math.h>

#define N_NODES 100000
#define E_EDGES 1600000
#define DHID    256
#define NHEAD   8
#define DHEAD   32
#define NEG_SLOPE 0.2f

typedef __attribute__((ext_vector_type(2))) float v2f;
typedef __attribute__((ext_vector_type(8))) float v8f;

// ---- monotone float<->uint key for atomic max over signed floats ----
__device__ __forceinline__ unsigned fkey(float f) {
    unsigned u = __float_as_uint(f);
    return (u & 0x80000000u) ? ~u : (u | 0x80000000u);
}
__device__ __forceinline__ float funkey(unsigned k) {
    unsigned u = (k & 0x80000000u) ? (k ^ 0x80000000u) : ~k;
    return __uint_as_float(u);
}
__device__ __forceinline__ float leaky(float v) {
    return v >= 0.0f ? v : NEG_SLOPE * v;
}

// =====================================================================
// K1: fused triple GEMM  out = x @ W (+bias)(+relu)   via fp32 WMMA
//     blockDim=256 (8 waves). Block covers 128 rows x 16 cols.
//     grid.z: 0 -> h=relu(x@Wh+bh), 1 -> xs=x@Wg, 2 -> xl=x@Wl+bl
//     B tile (256x16, 16KB) staged in LDS once per block.
// =====================================================================
__global__ __launch_bounds__(256)
void gemm_wmma_kernel(const float* __restrict__ x,
                      const float* __restrict__ Wh, const float* __restrict__ bh,
                      const float* __restrict__ Wg,
                      const float* __restrict__ Wl, const float* __restrict__ bl,
                      float* __restrict__ h_out, float* __restrict__ xs_out,
                      float* __restrict__ xl_out)
{
    __shared__ float lds_b[DHID * 16];

    const int which = blockIdx.z;
    const float* W    = (which == 0) ? Wh : (which == 1 ? Wg : Wl);
    const float* bias = (which == 0) ? bh : (which == 2 ? bl : nullptr);
    float* out        = (which == 0) ? h_out : (which == 1 ? xs_out : xl_out);
    const bool do_relu = (which == 0);

    const int col0 = blockIdx.y * 16;

    // cooperative stage of the 256x16 B tile into LDS
    for (int i = threadIdx.x; i < DHID * 16; i += 256) {
        int k = i >> 4, n = i & 15;
        lds_b[i] = W[k * DHID + col0 + n];
    }
    __syncthreads();

    const int lane = threadIdx.x & 31;       // wave32
    const int wave = threadIdx.x >> 5;
    const int row0 = blockIdx.x * 128 + wave * 16;
    const int m    = lane & 15;
    const int kb   = (lane >> 4) << 1;       // lanes 0-15: K=0,1 ; lanes 16-31: K=2,3

    int arow = row0 + m;
    if (arow >= N_NODES) arow = N_NODES - 1; // clamp: keep EXEC all-ones for WMMA
    const float* aptr = x + (size_t)arow * DHID + kb;

    v8f acc = {};
    #pragma unroll 8
    for (int k = 0; k < DHID; k += 4) {
        v2f a = *(const v2f*)(aptr + k);                 // A[m][k+kb .. k+kb+1]
        v2f b;
        b.x = lds_b[(k + kb)     * 16 + m];              // B[k+kb  ][col0+m]
        b.y = lds_b[(k + kb + 1) * 16 + m];              // B[k+kb+1][col0+m]
        acc = __builtin_amdgcn_wmma_f32_16x16x4_f32(
                  false, a, false, b, (short)0, acc, false, false);
    }

    // C/D layout: VGPR i -> M=i (lanes 0-15), M=i+8 (lanes 16-31); N = lane&15
    const int rbase = row0 + ((lane >> 4) << 3);
    const float bv = bias ? bias[col0 + m] : 0.0f;
    #pragma unroll
    for (int i = 0; i < 8; ++i) {
        int r = rbase + i;
        if (r < N_NODES) {
            float v = acc[i] + bv;
            if (do_relu) v = fmaxf(v, 0.0f);
            out[(size_t)r * DHID + col0 + m] = v;
        }
    }
}

// =====================================================================
// K2: zero gat accumulator, init segment-max keys and denominators
// =====================================================================
__global__ void init_kernel(float* __restrict__ gat,
                            unsigned* __restrict__ mkey,
                            float* __restrict__ denom)
{
    int i = blockIdx.x * blockDim.x + threadIdx.x;
    if (i < N_NODES * DHID) gat[i] = 0.0f;
    if (i < N_NODES * NHEAD) { mkey[i] = 0u; denom[i] = 0.0f; }
}

// =====================================================================
// K3: per-(node,head) attention scores  a_src/a_dst = <xs[n,h,:], att>
// =====================================================================
__global__ void attn_scores_kernel(const float* __restrict__ xs,
                                   const float* __restrict__ att_src,
                                   const float* __restrict__ att_dst,
                                   float* __restrict__ a_src,
                                   float* __restrict__ a_dst)
{
    int i = blockIdx.x * blockDim.x + threadIdx.x;
    if (i >= N_NODES * NHEAD) return;
    int n = i >> 3, h = i & 7;
    const float* xp = xs + (size_t)n * DHID + h * DHEAD;
    const float* as = att_src + h * DHEAD;
    const float* ad = att_dst + h * DHEAD;
    float s1 = 0.0f, s2 = 0.0f;
    #pragma unroll
    for (int d = 0; d < DHEAD; ++d) {
        float v = xp[d];
        s1 = fmaf(v, as[d], s1);
        s2 = fmaf(v, ad[d], s2);
    }
    a_src[i] = s1;
    a_dst[i] = s2;
}

// =====================================================================
// K4: segment max of leaky_relu scores via monotone-key atomicMax
// =====================================================================
__global__ void edge_max_kernel(const int* __restrict__ ei,
                                const float* __restrict__ a_src,
                                const float* __restrict__ a_dst,
                                unsigned* __restrict__ mkey)
{
    int e = blockIdx.x * blockDim.x + threadIdx.x;
    if (e >= E_EDGES) return;
    int s = ei[e], d = ei[E_EDGES + e];
    #pragma unroll
    for (int h = 0; h < NHEAD; ++h) {
        float v = leaky(a_src[s * NHEAD + h] + a_dst[d * NHEAD + h]);
        atomicMax(&mkey[d * NHEAD + h], fkey(v));
    }
}

// =====================================================================
// K5: m = isfinite(max) ? max : 0
// =====================================================================
__global__ void fix_m_kernel(const unsigned* __restrict__ mkey,
                             float* __restrict__ m)
{
    int i = blockIdx.x * blockDim.x + threadIdx.x;
    if (i >= N_NODES * NHEAD) return;
    float f = funkey(mkey[i]);
    m[i] = __builtin_isfinite(f) ? f : 0.0f;
}

// =====================================================================
// K6: denom = segment_sum( exp(e - m[dst]) * edge_attr )
// =====================================================================
__global__ void edge_sum_kernel(const int* __restrict__ ei,
                                const float* __restrict__ ea,
                                const float* __restrict__ a_src,
                                const float* __restrict__ a_dst,
                                const float* __restrict__ m,
                                float* __restrict__ denom)
{
    int e = blockIdx.x * blockDim.x + threadIdx.x;
    if (e >= E_EDGES) return;
    int s = ei[e], d = ei[E_EDGES + e];
    float w = ea[e];
    #pragma unroll
    for (int h = 0; h < NHEAD; ++h) {
        float v = leaky(a_src[s * NHEAD + h] + a_dst[d * NHEAD + h]);
        float p = __expf(v - m[d * NHEAD + h]) * w;
        atomicAdd(&denom[d * NHEAD + h], p);
    }
}

// =====================================================================
// K7: message scatter. One wave32 per edge; lane l of head j owns
//     channel j*32+l -> coalesced gather of xs[src] and contiguous
//     atomics into gat[dst] (102MB accumulator, L2-resident).
// =====================================================================
__global__ __launch_bounds__(256)
void edge_scatter_kernel(const int* __restrict__ ei,
                         const float* __restrict__ ea,
                         const float* __restrict__ a_src,
                         const float* __restrict__ a_dst,
                         const float* __restrict__ m,
                         const float* __restrict__ denom,
                         const float* __restrict__ xs,
                         float* __restrict__ gat)
{
    const int lane = threadIdx.x & 31;
    const int e = blockIdx.x * 8 + (threadIdx.x >> 5);
    if (e >= E_EDGES) return;
    const int s = ei[e], d = ei[E_EDGES + e];
    const float w = ea[e];
    const float* xsrc = xs + (size_t)s * DHID;
    float* gdst = gat + (size_t)d * DHID;
    #pragma unroll
    for (int h = 0; h < NHEAD; ++h) {
        float v = leaky(a_src[s * NHEAD + h] + a_dst[d * NHEAD + h]);
        float p = __expf(v - m[d * NHEAD + h]) * w;
        float alpha = p / (denom[d * NHEAD + h] + 1e-16f);
        atomicAdd(&gdst[h * DHEAD + lane], alpha * xsrc[h * DHEAD + lane]);
    }
}

// =====================================================================
// K8: fused epilogue. One wave32 per node, 8 channels/lane, wave-wide
//     shfl_xor reduction for LayerNorm stats.
//     out = (1-beta)*LN(h*xg) + beta*xg + x ,  xg = relu(gat + xl)
// =====================================================================
__global__ __launch_bounds__(256)
void final_kernel(const float* __restrict__ x,
                  const float* __restrict__ hbuf,
                  const float* __restrict__ xlbuf,
                  const float* __restrict__ gat,
                  const float* __restrict__ ln_g,
                  const float* __restrict__ ln_b,
                  const float* __restrict__ bparam,
                  float* __restrict__ out)
{
    const int lane = threadIdx.x & 31;
    const int n = blockIdx.x * 8 + (threadIdx.x >> 5);
    if (n >= N_NODES) return;
    const size_t base = (size_t)n * DHID;

    float t[8], xg[8];
    float sum = 0.0f, sumsq = 0.0f;
    #pragma unroll
    for (int j = 0; j < 8; ++j) {
        int c = lane + 32 * j;
        float g = fmaxf(gat[base + c] + xlbuf[base + c], 0.0f);
        xg[j] = g;
        float tv = hbuf[base + c] * g;
        t[j] = tv;
        sum += tv;
        sumsq = fmaf(tv, tv, sumsq);
    }
    #pragma unroll
    for (int o = 16; o >= 1; o >>= 1) {           // wave32 reduction
        sum   += __shfl_xor(sum,   o, 32);
        sumsq += __shfl_xor(sumsq, o, 32);
    }
    const float mu  = sum * (1.0f / DHID);
    const float var = sumsq * (1.0f / DHID) - mu * mu;
    const float inv = rsqrtf(var + 1e-5f);

    #pragma unroll
    for (int j = 0; j < 8; ++j) {
        int c = lane + 32 * j;
        float beta = bparam[c];
        float ln = (t[j] - mu) * inv * ln_g[c] + ln_b[c];
        out[base + c] = (1.0f - beta) * ln + beta * xg[j] + x[base + c];
    }
}

// =====================================================================
extern "C" void kernel_launch(void* const* d_in, const int* in_sizes, int n_in,
                              void* d_out, int out_size, void* d_ws, size_t ws_size,
                              hipStream_t stream)
{
    (void)in_sizes; (void)n_in; (void)out_size; (void)ws_size;

    const float* x        = (const float*)d_in[0];
    const int*   ei       = (const int*)  d_in[1];   // [2,E]: row0=src, row1=dst
    const float* ea       = (const float*)d_in[2];
    const float* Wg       = (const float*)d_in[3];
    const float* att_src  = (const float*)d_in[4];
    const float* att_dst  = (const float*)d_in[5];
    const float* Wh       = (const float*)d_in[6];
    const float* bh       = (const float*)d_in[7];
    const float* Wl       = (const float*)d_in[8];
    const float* bl       = (const float*)d_in[9];
    const float* ln_g     = (const float*)d_in[10];
    const float* ln_b     = (const float*)d_in[11];
    const float* b_param  = (const float*)d_in[12];
    float* out            = (float*)d_out;

    // ---- workspace layout (floats) ----
    const size_t NB  = (size_t)N_NODES * DHID;   // 25.6M
    const size_t NHf = (size_t)N_NODES * NHEAD;  // 0.8M
    float*    xs    = (float*)d_ws;
    float*    hb    = xs + NB;
    float*    xl    = hb + NB;
    float*    gat   = xl + NB;
    float*    a_src = gat + NB;
    float*    a_dst = a_src + NHf;
    unsigned* mkey  = (unsigned*)(a_dst + NHf);
    float*    m     = (float*)(mkey + NHf);
    float*    denom = m + NHf;

    // K1: three GEMMs (z selects matrix); 782*128 >= 100000 rows, 16 col tiles
    dim3 gemm_grid((N_NODES + 127) / 128, DHID / 16, 3);
    gemm_wmma_kernel<<<gemm_grid, 256, 0, stream>>>(x, Wh, bh, Wg, Wl, bl, hb, xs, xl);

    // K2: init accumulators
    init_kernel<<<(N_NODES * DHID + 255) / 256, 256, 0, stream>>>(gat, mkey, denom);

    // K3: attention scores (needs xs)
    attn_scores_kernel<<<(N_NODES * NHEAD + 255) / 256, 256, 0, stream>>>(
        xs, att_src, att_dst, a_src, a_dst);

    // K4: segment max
    edge_max_kernel<<<(E_EDGES + 255) / 256, 256, 0, stream>>>(ei, a_src, a_dst, mkey);

    // K5: finite fixup
    fix_m_kernel<<<(N_NODES * NHEAD + 255) / 256, 256, 0, stream>>>(mkey, m);

    // K6: softmax denominators
    edge_sum_kernel<<<(E_EDGES + 255) / 256, 256, 0, stream>>>(ei, ea, a_src, a_dst, m, denom);

    // K7: message scatter, one wave per edge
    edge_scatter_kernel<<<(E_EDGES + 7) / 8, 256, 0, stream>>>(
        ei, ea, a_src, a_dst, m, denom, xs, gat);

    // K8: fused epilogue, one wave per node
    final_kernel<<<(N_NODES + 7) / 8, 256, 0, stream>>>(
        x, hb, xl, gat, ln_g, ln_b, b_param, out);
}